// ResidualBlock_56599079027000
// MI455X (gfx1250) — compile-verified
//
#include <hip/hip_runtime.h>
#include <hip/hip_bf16.h>

typedef __attribute__((ext_vector_type(2))) float v2f;
typedef __attribute__((ext_vector_type(8))) float v8f;

#define GEPS 1e-15f
#define BNEPS 1e-5f

// ---------------------------------------------------------------------------
// Zero fill
// ---------------------------------------------------------------------------
__global__ void zero_f32(float* __restrict__ p, int n) {
    int i = blockIdx.x * blockDim.x + threadIdx.x;
    if (i < n) p[i] = 0.0f;
}

// ---------------------------------------------------------------------------
// C[M, ncols] = A[M, 32] @ B[32, ncols] using V_WMMA_F32_16X16X4_F32.
// One wave32 per 16x16 output tile. M % 16 == 0, ncols % 16 == 0 (true here:
// M = 50000, ncols in {160, 32}), so EXEC is all-ones for every WMMA wave.
// A-layout (16x4 f32): lane = {A[m][k0], A[m][k0+1]}, m = lane&15, k0 = 2*(lane>>4)
// B-layout (4x16 f32): lane = {B[k0][n],  B[k0+1][n]},  n = lane&15
// D-layout (16x16 f32): vgpr v, lane l -> D[v + 8*(l>>4)][l&15]
// ---------------------------------------------------------------------------
__global__ void gemm32_wmma(const float* __restrict__ A, const float* __restrict__ B,
                            float* __restrict__ C, int M, int ncols) {
    int wave = (blockIdx.x * blockDim.x + threadIdx.x) >> 5;
    int lane = threadIdx.x & 31;
    int colTiles = ncols >> 4;
    int totalTiles = (M >> 4) * colTiles;
    if (wave >= totalTiles) return;                 // wave-uniform exit
    int rowTile = wave / colTiles;
    int colTile = wave - rowTile * colTiles;

    int m   = lane & 15;
    int kp  = (lane >> 4) << 1;                     // 0 or 2
    int row = rowTile * 16 + m;
    int col = colTile * 16 + m;                     // B column (n = lane&15)

    v8f acc = {};
#pragma unroll
    for (int kb = 0; kb < 8; ++kb) {
        int k0 = kb * 4 + kp;
        v2f a = *(const v2f*)(A + row * 32 + k0);   // 8B-aligned pair
        v2f b;
        b.x = B[k0 * ncols + col];
        b.y = B[(k0 + 1) * ncols + col];
        acc = __builtin_amdgcn_wmma_f32_16x16x4_f32(
            /*neg_a=*/false, a, /*neg_b=*/false, b,
            /*c_mod=*/(short)0, acc, /*reuse_a=*/false, /*reuse_b=*/false);
    }
    int mBase = rowTile * 16 + ((lane >> 4) << 3);
    int n     = colTile * 16 + (lane & 15);
#pragma unroll
    for (int v = 0; v < 8; ++v)
        C[(mBase + v) * ncols + n] = acc[v];
}

// ---------------------------------------------------------------------------
// Gaussian mixture weight helper
// ---------------------------------------------------------------------------
__device__ __forceinline__ float gmm_w(float a0, float a1, float a2,
                                       const float* __restrict__ mu,
                                       const float* __restrict__ sig, int k) {
    float d0 = a0 - mu[k * 3 + 0], s0 = sig[k * 3 + 0];
    float d1 = a1 - mu[k * 3 + 1], s1 = sig[k * 3 + 1];
    float d2 = a2 - mu[k * 3 + 2], s2 = sig[k * 3 + 2];
    float q = d0 * d0 / (GEPS + s0 * s0)
            + d1 * d1 / (GEPS + s1 * s1)
            + d2 * d2 / (GEPS + s2 * s2);
    return expf(-0.5f * q);
}

// ---------------------------------------------------------------------------
// Edge pass 1: fused conv1 (K=5) + skip conv (K=1) scatter, plus degree count.
// One wave32 per edge; lane == output channel (coalesced 128B per k-slice).
// ---------------------------------------------------------------------------
__global__ void edge_pass1(const long long* __restrict__ ei, const float* __restrict__ ea,
                           const float* __restrict__ mu1, const float* __restrict__ sig1,
                           const float* __restrict__ mus, const float* __restrict__ sigs,
                           const float* __restrict__ y1, const float* __restrict__ ys,
                           float* __restrict__ agg1, float* __restrict__ aggs,
                           float* __restrict__ cnt, int E) {
    int t = blockIdx.x * blockDim.x + threadIdx.x;
    int e = t >> 5;
    int c = t & 31;
    if (e >= E) return;
    long long src = ei[e];
    long long dst = ei[(long long)E + e];
    float a0 = ea[e * 3 + 0], a1 = ea[e * 3 + 1], a2 = ea[e * 3 + 2];

    float msg = 0.0f;
    const float* yb = y1 + src * 160 + c;
#pragma unroll
    for (int k = 0; k < 5; ++k)
        msg += gmm_w(a0, a1, a2, mu1, sig1, k) * yb[k * 32];

    float msgs = gmm_w(a0, a1, a2, mus, sigs, 0) * ys[src * 32 + c];

    atomicAdd(&agg1[dst * 32 + c], msg);
    atomicAdd(&aggs[dst * 32 + c], msgs);
    if (c == 0) atomicAdd(&cnt[dst], 1.0f);
}

// Edge pass 2: conv2 (K=5) only
__global__ void edge_pass2(const long long* __restrict__ ei, const float* __restrict__ ea,
                           const float* __restrict__ mu2, const float* __restrict__ sig2,
                           const float* __restrict__ y2, float* __restrict__ agg2, int E) {
    int t = blockIdx.x * blockDim.x + threadIdx.x;
    int e = t >> 5;
    int c = t & 31;
    if (e >= E) return;
    long long src = ei[e];
    long long dst = ei[(long long)E + e];
    float a0 = ea[e * 3 + 0], a1 = ea[e * 3 + 1], a2 = ea[e * 3 + 2];
    float msg = 0.0f;
    const float* yb = y2 + src * 160 + c;
#pragma unroll
    for (int k = 0; k < 5; ++k)
        msg += gmm_w(a0, a1, a2, mu2, sig2, k) * yb[k * 32];
    atomicAdd(&agg2[dst * 32 + c], msg);
}

// ---------------------------------------------------------------------------
// pre[i,c] = agg[i,c]/max(cnt[i],1) + xr[i,c] + bias[c]; accumulate per-channel
// sum / sumsq into stats[0..31] / stats[32..63] (global f32 atomics via LDS).
// ---------------------------------------------------------------------------
__global__ void combine_stats(const float* __restrict__ agg, const float* __restrict__ cnt,
                              const float* __restrict__ xr, const float* __restrict__ bias,
                              float* __restrict__ pre, float* __restrict__ stats, int N) {
    __shared__ float ssum[32], ssq[32];
    int t = threadIdx.x;
    if (t < 32) { ssum[t] = 0.0f; ssq[t] = 0.0f; }
    __syncthreads();
    int c = t & 31;                       // stride is a multiple of 32 -> channel fixed
    float b = bias[c];
    float ls = 0.0f, lq = 0.0f;
    int total = N * 32;
    for (int idx = blockIdx.x * blockDim.x + t; idx < total; idx += gridDim.x * blockDim.x) {
        int i = idx >> 5;
        float v = agg[idx] / fmaxf(cnt[i], 1.0f) + xr[idx] + b;
        pre[idx] = v;
        ls += v;
        lq += v * v;
    }
    atomicAdd(&ssum[c], ls);
    atomicAdd(&ssq[c], lq);
    __syncthreads();
    if (t < 32) {
        atomicAdd(&stats[t], ssum[t]);
        atomicAdd(&stats[32 + t], ssq[t]);
    }
}

__device__ __forceinline__ float eluf(float v) {
    return v > 0.0f ? v : (expf(v) - 1.0f);
}

// h = elu(bn(pre))
__global__ void bn_elu(const float* __restrict__ pre, const float* __restrict__ stats,
                       const float* __restrict__ gamma, const float* __restrict__ beta,
                       float* __restrict__ out, int N) {
    int idx = blockIdx.x * blockDim.x + threadIdx.x;
    if (idx >= N * 32) return;
    int c = idx & 31;
    float invN = 1.0f / (float)N;
    float mean = stats[c] * invN;
    float var  = stats[32 + c] * invN - mean * mean;
    float rs   = rsqrtf(var + BNEPS);
    out[idx] = eluf(gamma[c] * (pre[idx] - mean) * rs + beta[c]);
}

// out = elu(bn(pre2) + bn(pres))
__global__ void final_out(const float* __restrict__ pre2, const float* __restrict__ st2,
                          const float* __restrict__ gam2, const float* __restrict__ bet2,
                          const float* __restrict__ pres, const float* __restrict__ sts,
                          const float* __restrict__ gams, const float* __restrict__ bets,
                          float* __restrict__ out, int N) {
    int idx = blockIdx.x * blockDim.x + threadIdx.x;
    if (idx >= N * 32) return;
    int c = idx & 31;
    float invN = 1.0f / (float)N;

    float m2 = st2[c] * invN;
    float v2 = st2[32 + c] * invN - m2 * m2;
    float h  = gam2[c] * (pre2[idx] - m2) * rsqrtf(v2 + BNEPS) + bet2[c];

    float ms = sts[c] * invN;
    float vs = sts[32 + c] * invN - ms * ms;
    float s  = gams[c] * (pres[idx] - ms) * rsqrtf(vs + BNEPS) + bets[c];

    out[idx] = eluf(h + s);
}

// ---------------------------------------------------------------------------
extern "C" void kernel_launch(void* const* d_in, const int* in_sizes, int n_in,
                              void* d_out, int out_size, void* d_ws, size_t ws_size,
                              hipStream_t stream) {
    const float* x     = (const float*)d_in[0];
    const float* ea    = (const float*)d_in[1];
    const float* g1    = (const float*)d_in[2];
    const float* mu1   = (const float*)d_in[3];
    const float* sig1  = (const float*)d_in[4];
    const float* root1 = (const float*)d_in[5];
    const float* b1    = (const float*)d_in[6];
    const float* gam1  = (const float*)d_in[7];
    const float* bet1  = (const float*)d_in[8];
    const float* g2    = (const float*)d_in[9];
    const float* mu2   = (const float*)d_in[10];
    const float* sig2  = (const float*)d_in[11];
    const float* root2 = (const float*)d_in[12];
    const float* b2    = (const float*)d_in[13];
    const float* gam2  = (const float*)d_in[14];
    const float* bet2  = (const float*)d_in[15];
    const float* gs    = (const float*)d_in[16];
    const float* mus   = (const float*)d_in[17];
    const float* sigs  = (const float*)d_in[18];
    const float* roots = (const float*)d_in[19];
    const float* bs    = (const float*)d_in[20];
    const float* gams  = (const float*)d_in[21];
    const float* bets  = (const float*)d_in[22];
    const long long* ei = (const long long*)d_in[23];

    const int N = in_sizes[0] / 32;
    const int E = in_sizes[23] / 2;

    // Workspace layout (floats)
    float* w     = (float*)d_ws;
    float* cnt   = w;                 w += N;        // degree
    float* y     = w;                 w += N * 160;  // y1, reused as y2
    float* ys    = w;                 w += N * 32;   // x @ gs
    float* xr    = w;                 w += N * 32;   // x@root1, reused as h@root2
    float* xrs   = w;                 w += N * 32;   // x@roots
    float* agg   = w;                 w += N * 32;   // agg1, reused as agg2
    float* aggs  = w;                 w += N * 32;   // skip aggregation
    float* pre1  = w;                 w += N * 32;   // pre-BN 1, reused as pre-BN 2
    float* h     = w;                 w += N * 32;   // elu(bn1)
    float* pres  = w;                 w += N * 32;   // skip pre-BN
    float* stats = w;                 /* 3 * 64 */   // [st1 | sts | st2]

    const int TPB = 256;
    auto zero = [&](float* p, int n) {
        zero_f32<<<(n + TPB - 1) / TPB, TPB, 0, stream>>>(p, n);
    };
    auto gemm = [&](const float* A, const float* B, float* C, int ncols) {
        int tiles = (N >> 4) * (ncols >> 4);
        gemm32_wmma<<<(tiles + 7) / 8, TPB, 0, stream>>>(A, B, C, N, ncols);
    };

    // init accumulators
    zero(agg, N * 32);
    zero(aggs, N * 32);
    zero(cnt, N);
    zero(stats, 3 * 64);

    // node projections for conv1 + skip (WMMA GEMMs)
    gemm(x, g1, y, 160);
    gemm(x, gs, ys, 32);
    gemm(x, root1, xr, 32);
    gemm(x, roots, xrs, 32);

    // fused edge scatter: conv1 + skip (+ degree)
    edge_pass1<<<(E * 32 + TPB - 1) / TPB, TPB, 0, stream>>>(
        ei, ea, mu1, sig1, mus, sigs, y, ys, agg, aggs, cnt, E);

    // pre-activation + BN stats
    combine_stats<<<960, TPB, 0, stream>>>(agg, cnt, xr, b1, pre1, stats, N);
    combine_stats<<<960, TPB, 0, stream>>>(aggs, cnt, xrs, bs, pres, stats + 64, N);

    // h = elu(bn1(pre1))
    bn_elu<<<(N * 32 + TPB - 1) / TPB, TPB, 0, stream>>>(pre1, stats, gam1, bet1, h, N);

    // conv2: reuse agg / y / xr buffers
    zero(agg, N * 32);
    gemm(h, g2, y, 160);
    gemm(h, root2, xr, 32);

    edge_pass2<<<(E * 32 + TPB - 1) / TPB, TPB, 0, stream>>>(
        ei, ea, mu2, sig2, y, agg, E);

    combine_stats<<<960, TPB, 0, stream>>>(agg, cnt, xr, b2, pre1, stats + 128, N);

    // out = elu(bn2(pre2) + bns(pres))
    final_out<<<(N * 32 + TPB - 1) / TPB, TPB, 0, stream>>>(
        pre1, stats + 128, gam2, bet2, pres, stats + 64, gams, bets,
        (float*)d_out, N);
}